// MultiViewModel_20624432956098
// MI455X (gfx1250) — compile-verified
//
#include <hip/hip_runtime.h>
#include <hip/hip_bf16.h>

// ---------------------------------------------------------------------------
// MultiViewModel on MI455X (gfx1250, wave32).
// Kernels: init (zero scalar slots) -> line losses (gather+dot, atomic acc)
//          -> attention fusion (WMMA f32 16x16x4 GEMM for score(X)=lrelu(XW+b)v)
//          -> supervised loss (gather+dot over fused_emb).
// d_out layout: [0] = loss_uu+loss_up+loss_pp, [1..32000000] = fused_emb,
//               [32000001] = loss_supervised.
// ---------------------------------------------------------------------------

typedef __attribute__((ext_vector_type(2))) float v2f;
typedef __attribute__((ext_vector_type(8))) float v8f;

#define EMBED_DIM 128
#define SIZE_UU   50000
#define SIZE_UP   250000
#define B_EDGES   100000
#define K_NEG     5
#define N_USERS   20000
#define H_HIST    50
#define N_TILES   (SIZE_UP / 16)   // 15625
#define XSTRIDE   132              // padded row stride in LDS (floats)
#define WAVES_PB  8                // waves per block (256 threads)

// ---- wave32 reductions ----------------------------------------------------
__device__ __forceinline__ float wred32(float x) {   // full-wave sum
    x += __shfl_xor(x, 1, 32);
    x += __shfl_xor(x, 2, 32);
    x += __shfl_xor(x, 4, 32);
    x += __shfl_xor(x, 8, 32);
    x += __shfl_xor(x, 16, 32);
    return x;
}
__device__ __forceinline__ float hred16(float x) {   // sum within 16-lane half
    x += __shfl_xor(x, 1, 32);
    x += __shfl_xor(x, 2, 32);
    x += __shfl_xor(x, 4, 32);
    x += __shfl_xor(x, 8, 32);
    return x;
}
__device__ __forceinline__ float log_sig(float x) {  // jax.nn.log_sigmoid
    return fminf(x, 0.f) - log1pf(__expf(-fabsf(x)));
}
__device__ __forceinline__ float dot4(float4 a, float4 b) {
    return a.x * b.x + a.y * b.y + a.z * b.z + a.w * b.w;
}

// ---- init: zero the two scalar accumulator slots ---------------------------
__global__ void init_kernel(float* __restrict__ out, int out_size) {
    if (threadIdx.x == 0 && blockIdx.x == 0) {
        out[0] = 0.f;
        out[out_size - 1] = 0.f;
    }
}

// ---- LINE losses: one wave per edge, 3 tables ------------------------------
__global__ __launch_bounds__(256) void line_loss_kernel(
    const float* __restrict__ emb_uu, const float* __restrict__ emb_up,
    const float* __restrict__ emb_pp,
    const int* __restrict__ vi_uu, const int* __restrict__ vj_uu, const int* __restrict__ ng_uu,
    const int* __restrict__ vi_up, const int* __restrict__ vj_up, const int* __restrict__ ng_up,
    const int* __restrict__ vi_pp, const int* __restrict__ vj_pp, const int* __restrict__ ng_pp,
    float* __restrict__ loss_out)
{
    const int g    = (blockIdx.x * blockDim.x + threadIdx.x) >> 5;  // wave id
    const int lane = threadIdx.x & 31;
    if (g >= 3 * B_EDGES) return;                       // wave-uniform exit

    const int table = g / B_EDGES;
    const int e     = g - table * B_EDGES;
    const float* emb; const int *vi, *vj, *ng;
    if (table == 0)      { emb = emb_uu; vi = vi_uu; vj = vj_uu; ng = ng_uu; }
    else if (table == 1) { emb = emb_up; vi = vi_up; vj = vj_up; ng = ng_up; }
    else                 { emb = emb_pp; vi = vi_pp; vj = vj_pp; ng = ng_pp; }

    const int i = vi[e], j = vj[e];
    const float4 ei = *(const float4*)(emb + (size_t)i * EMBED_DIM + lane * 4);
    const float4 ej = *(const float4*)(emb + (size_t)j * EMBED_DIM + lane * 4);

    float acc = log_sig(wred32(dot4(ei, ej)));          // positive term
    #pragma unroll
    for (int k = 0; k < K_NEG; ++k) {
        const int n = ng[(size_t)e * K_NEG + k];
        const float4 en = *(const float4*)(emb + (size_t)n * EMBED_DIM + lane * 4);
        acc += log_sig(-wred32(dot4(ei, en)));          // e_n = -emb[neg]
    }
    if (lane == 0) atomicAdd(loss_out, -acc * (1.0f / (float)B_EDGES));
}

// ---- WMMA GEMM: H = leaky_relu(X@W + b); sp[j] = row-score partials --------
// xs: 16 rows (stride XSTRIDE) in LDS.  A-frag layout (f32 16x16x4):
//   lanes 0-15 hold K=4k,4k+1 for M=lane; lanes 16-31 hold K=4k+2,4k+3.
// C/D layout: VGPR j, lanes 0-15 -> M=j, lanes 16-31 -> M=j+8; N = lane&15 (+16*nt).
__device__ __forceinline__ void gemm_score(
    const float* __restrict__ xs, const float* __restrict__ W,
    const float* __restrict__ bias, const float* __restrict__ vvec,
    int cl, int half, float sp[8])
{
    #pragma unroll
    for (int j = 0; j < 8; ++j) sp[j] = 0.f;

    for (int nt = 0; nt < 8; ++nt) {
        const int col = nt * 16 + cl;
        v8f acc = {};
        for (int k = 0; k < 32; ++k) {
            const int kk = 4 * k + 2 * half;
            v2f a = *(const v2f*)(xs + cl * XSTRIDE + kk);   // 8B-aligned LDS
            v2f b;
            b.x = W[kk * EMBED_DIM + col];                   // L2-hot, coalesced
            b.y = W[(kk + 1) * EMBED_DIM + col];
            acc = __builtin_amdgcn_wmma_f32_16x16x4_f32(
                false, a, false, b, (short)0, acc, false, false);
        }
        const float bc = bias[col];
        const float vc = vvec[col];
        #pragma unroll
        for (int j = 0; j < 8; ++j) {
            float h = acc[j] + bc;
            h = (h > 0.f) ? h : 0.01f * h;                   // leaky_relu(0.01)
            sp[j] += h * vc;
        }
    }
    #pragma unroll
    for (int j = 0; j < 8; ++j) sp[j] = hred16(sp[j]);       // sum over N
}

// ---- attention fusion: one wave per 16-row tile ----------------------------
__global__ __launch_bounds__(256) void fuse_kernel(
    const float* __restrict__ emb_uu, const float* __restrict__ emb_up,
    const float* __restrict__ emb_pp, const float* __restrict__ W,
    const float* __restrict__ bias, const float* __restrict__ vvec,
    float* __restrict__ out /* = d_out + 1, 4B-aligned only */)
{
    __shared__ float XS[WAVES_PB][2][16 * XSTRIDE];   // ~135 KB
    const int lane = threadIdx.x & 31;
    const int wid  = threadIdx.x >> 5;
    const int tile = blockIdx.x * WAVES_PB + wid;
    if (tile >= N_TILES) return;                      // wave-uniform exit

    const int row0 = tile * 16;
    float* xa = &XS[wid][0][0];
    float* xb = &XS[wid][1][0];

    // Stage 16 A-rows + 16 B-rows (coalesced 512B per row).
    for (int m = 0; m < 16; ++m) {
        const int r = row0 + m;
        const float* pa = (r == 0) ? emb_pp
                        : (r < SIZE_UU ? emb_uu + (size_t)r * EMBED_DIM
                                       : emb_pp + (size_t)(r - 1) * EMBED_DIM);
        const float* pb = (r == 0) ? emb_pp : emb_up + (size_t)r * EMBED_DIM;
        *(float4*)(xa + m * XSTRIDE + lane * 4) = *(const float4*)(pa + lane * 4);
        *(float4*)(xb + m * XSTRIDE + lane * 4) = *(const float4*)(pb + lane * 4);
    }

    const int cl = lane & 15, half = lane >> 4;
    float spA[8], spB[8];
    gemm_score(xa, W, bias, vvec, cl, half, spA);
    gemm_score(xb, W, bias, vvec, cl, half, spB);

    // Per-row softmax over {sA, sB} and fused output (b32 stores: out is +1 float).
    #pragma unroll
    for (int m = 0; m < 16; ++m) {
        const float sa = __shfl(spA[m & 7], (m >> 3) * 16, 32);
        const float sb = __shfl(spB[m & 7], (m >> 3) * 16, 32);
        const float mx = fmaxf(sa, sb);
        const float ea = __expf(sa - mx), eb = __expf(sb - mx);
        const float inv = 1.f / (ea + eb);
        const float a0 = ea * inv, a1 = eb * inv;
        const size_t base = (size_t)(row0 + m) * EMBED_DIM;
        for (int c = lane; c < EMBED_DIM; c += 32)
            out[base + c] = xa[m * XSTRIDE + c] * a0 + xb[m * XSTRIDE + c] * a1;
    }
}

// ---- supervised loss: one wave per user ------------------------------------
__global__ __launch_bounds__(256) void supervised_kernel(
    const float* __restrict__ fused /* d_out+1 */,
    const int* __restrict__ user_ids, const int* __restrict__ user_hist,
    const int* __restrict__ hist_len, float* __restrict__ loss_out)
{
    const int g    = (blockIdx.x * blockDim.x + threadIdx.x) >> 5;
    const int lane = threadIdx.x & 31;
    if (g >= N_USERS) return;                          // wave-uniform exit

    const int uid = user_ids[g];
    const size_t ub = (size_t)uid * EMBED_DIM;
    // fused is only 4B-aligned -> scalar b32 loads (still coalesced).
    float u0 = fused[ub + lane], u1 = fused[ub + lane + 32];
    float u2 = fused[ub + lane + 64], u3 = fused[ub + lane + 96];

    const int hl = hist_len[g];
    float acc = 0.f;
    for (int h = 0; h < H_HIST; ++h) {
        if (h < hl) {
            const int pid = user_hist[g * H_HIST + h];
            const size_t pb = (size_t)pid * EMBED_DIM;
            acc += fused[pb + lane]      * u0 + fused[pb + lane + 32] * u1
                 + fused[pb + lane + 64] * u2 + fused[pb + lane + 96] * u3;
        }
    }
    acc = wred32(acc);
    if (lane == 0) atomicAdd(loss_out, -acc);
}

// ---------------------------------------------------------------------------
extern "C" void kernel_launch(void* const* d_in, const int* in_sizes, int n_in,
                              void* d_out, int out_size, void* d_ws, size_t ws_size,
                              hipStream_t stream) {
    (void)in_sizes; (void)n_in; (void)d_ws; (void)ws_size;
    const float* emb_uu = (const float*)d_in[0];
    const float* emb_up = (const float*)d_in[1];
    const float* emb_pp = (const float*)d_in[2];
    const float* att_W  = (const float*)d_in[3];
    const float* att_b  = (const float*)d_in[4];
    const float* att_v  = (const float*)d_in[5];
    const int* vi_uu = (const int*)d_in[6];
    const int* vj_uu = (const int*)d_in[7];
    const int* ng_uu = (const int*)d_in[8];
    const int* vi_up = (const int*)d_in[9];
    const int* vj_up = (const int*)d_in[10];
    const int* ng_up = (const int*)d_in[11];
    const int* vi_pp = (const int*)d_in[12];
    const int* vj_pp = (const int*)d_in[13];
    const int* ng_pp = (const int*)d_in[14];
    const int* user_ids  = (const int*)d_in[15];
    const int* user_hist = (const int*)d_in[16];
    const int* hist_len  = (const int*)d_in[17];
    float* out = (float*)d_out;

    init_kernel<<<1, 32, 0, stream>>>(out, out_size);

    // 3 * 100000 edge-waves, 8 waves per 256-thread block.
    line_loss_kernel<<<(3 * B_EDGES + WAVES_PB - 1) / WAVES_PB, 256, 0, stream>>>(
        emb_uu, emb_up, emb_pp,
        vi_uu, vj_uu, ng_uu, vi_up, vj_up, ng_up, vi_pp, vj_pp, ng_pp, out);

    // 15625 tiles, one per wave.
    fuse_kernel<<<(N_TILES + WAVES_PB - 1) / WAVES_PB, 256, 0, stream>>>(
        emb_uu, emb_up, emb_pp, att_W, att_b, att_v, out + 1);

    // 20000 user-waves; reads fused_emb written by fuse_kernel (same stream).
    supervised_kernel<<<(N_USERS + WAVES_PB - 1) / WAVES_PB, 256, 0, stream>>>(
        out + 1, user_ids, user_hist, hist_len, out + out_size - 1);
}